// conGraphConvolutionlayer_76012331205188
// MI455X (gfx1250) — compile-verified
//
#include <hip/hip_runtime.h>
#include <hip/hip_bf16.h>
#include <stdint.h>

typedef __attribute__((ext_vector_type(2))) float    v2f;
typedef __attribute__((ext_vector_type(8))) float    v8f;
typedef __attribute__((ext_vector_type(4))) uint32_t u32x4;
typedef __attribute__((ext_vector_type(8))) uint32_t u32x8;

#define NROWS 8192
#define NF    256
#define GAMMA_P 0.01f

#define MT     32          // M rows per workgroup
#define KSTEP  32          // K chunk staged in LDS
#define AP     36          // padded A-tile LDS stride (TDM pad: +4 DWORDs per 32-DWORD row)

// ---------------------------------------------------------------------------
// TDM: issue a 2D tile DMA (global -> LDS). D# groups per CDNA5 ISA 8.3/8.4.
// data_size = 4 bytes; tensor dims == tile dims (tiles always in bounds),
// tensor_dim0_stride carries the real row stride.
// ---------------------------------------------------------------------------
__device__ __forceinline__ void tdm_load_2d(uint32_t lds_off, const void* gaddr,
                                            uint32_t tile_w, uint32_t tile_h,
                                            uint64_t stride_elems, uint32_t pad_bits) {
  const uint64_t ga = (uint64_t)gaddr;
  u32x4 g0;
  g0[0] = 1u;                                            // count=1, user descriptor
  g0[1] = lds_off;                                       // lds_addr (bytes)
  g0[2] = (uint32_t)ga;                                  // global_addr[31:0]
  g0[3] = (uint32_t)((ga >> 32) & 0x01FFFFFFu) | (2u << 30);  // addr[56:32] | type=2
  u32x8 g1;
  g1[0] = (2u << 16) | pad_bits;                         // wg_mask=0, data_size=4B, pad cfg
  g1[1] = (tile_w & 0xFFFFu) << 16;                      // tensor_dim0[15:0] (bits 79:48)
  g1[2] = (tile_w >> 16) | ((tile_h & 0xFFFFu) << 16);   // tensor_dim0[31:16] | tensor_dim1[15:0]
  g1[3] = (tile_h >> 16) | (tile_w << 16);               // tensor_dim1[31:16] | tile_dim0
  g1[4] = tile_h & 0xFFFFu;                              // tile_dim1 | tile_dim2=0
  g1[5] = (uint32_t)(stride_elems & 0xFFFFFFFFu);        // tensor_dim0_stride[31:0]
  g1[6] = (uint32_t)((stride_elems >> 32) & 0xFFFFu);    // stride[47:32] | dim1_stride[15:0]=0
  g1[7] = 0u;                                            // dim1_stride[47:16]=0
  asm volatile("tensor_load_to_lds %0, %1" :: "s"(g0), "s"(g1) : "memory");
}

#define A_PAD_BITS ((1u << 20) | (4u << 22) | (3u << 25)) // pad_enable, every 32 DW, +4 DW

// ---------------------------------------------------------------------------
// Kernel 1: normalized Chebyshev weights  Wn_k = W_k + gamma * ||W_k||_F
// ---------------------------------------------------------------------------
__global__ void prep_weights_kernel(const float* __restrict__ W,
                                    float* __restrict__ Wn) {
  __shared__ float red[256];
  const int k = blockIdx.x;                 // 0..2
  const float* Wk  = W  + (size_t)k * NF * NF;
  float*       Wnk = Wn + (size_t)k * NF * NF;
  const int t = threadIdx.x;
  float s = 0.f;
  for (int i = t; i < NF * NF; i += 256) { float w = Wk[i]; s += w * w; }
  red[t] = s;
  __syncthreads();
  for (int off = 128; off > 0; off >>= 1) {
    if (t < off) red[t] += red[t + off];
    __syncthreads();
  }
  const float add = GAMMA_P * sqrtf(red[0]);
  for (int i = t; i < NF * NF; i += 256) Wnk[i] = Wk[i] + add;
}

// ---------------------------------------------------------------------------
// Kernel 2: d_inv_sqrt[i] = rsqrt(clip(sum_j adj[i][j], 1e-6))
// ---------------------------------------------------------------------------
__global__ void degree_kernel(const float* __restrict__ adj,
                              float* __restrict__ dinv) {
  __shared__ float red[256];
  const int row = blockIdx.x;
  const float* a = adj + (size_t)row * NROWS;
  const int t = threadIdx.x;
  float s = 0.f;
  for (int c = t; c < NROWS; c += 256) s += a[c];
  red[t] = s;
  __syncthreads();
  for (int off = 128; off > 0; off >>= 1) {
    if (t < off) red[t] += red[t + off];
    __syncthreads();
  }
  if (t == 0) dinv[row] = rsqrtf(fmaxf(red[0], 1e-6f));
}

// ---------------------------------------------------------------------------
// Kernel 3: out[j,:] = d[j] * in[j,:]
// ---------------------------------------------------------------------------
__global__ void scale_rows_kernel(const float* __restrict__ in,
                                  const float* __restrict__ d,
                                  float* __restrict__ out) {
  const size_t i4 = (size_t)blockIdx.x * blockDim.x + threadIdx.x; // one float4
  const int row = (int)((i4 * 4) >> 8);                            // /NF
  const float dv = d[row];
  float4 v = ((const float4*)in)[i4];
  v.x *= dv; v.y *= dv; v.z *= dv; v.w *= dv;
  ((float4*)out)[i4] = v;
}

// ---------------------------------------------------------------------------
// Kernel 4: f32 WMMA GEMM  acc = A[M x K] @ B[K x 256], TDM-fed LDS tiles,
// double buffered on TENSORcnt.
//   mode 0: out = acc            mode 1: out = -d[row]*acc
//   mode 2: out = -2*d[row]*acc - xorig      mode 3: out += acc
// ---------------------------------------------------------------------------
__global__ void __launch_bounds__(256)
gemm_wmma_kernel(const float* __restrict__ A, int lda, int K,
                 const float* __restrict__ B,
                 float* __restrict__ out,
                 const float* __restrict__ dinv,
                 const float* __restrict__ xorig,
                 int mode) {
  __shared__ float As[2 * MT * AP];        // 9216 B  (TDM-padded rows, stride 36)
  __shared__ float Bs[2 * KSTEP * NF];     // 65536 B (unpadded: frag reads are row-wise)

  const int t     = threadIdx.x;
  const int wave  = t >> 5;
  const int lane  = t & 31;
  const int l15   = lane & 15;
  const int khalf = lane >> 4;          // 0: K{0,1}, 1: K{2,3} of each WMMA step
  const int mt    = wave >> 2;          // 0..1   -> 16-row half
  const int ntb   = (wave & 3) * 4;     // 0,4,8,12 -> base 16-col tile
  const int m0    = blockIdx.x * MT;

  const uint32_t as_base = (uint32_t)(uintptr_t)&As[0];
  const uint32_t bs_base = (uint32_t)(uintptr_t)&Bs[0];
  const int nchunks = K / KSTEP;

  v8f acc0 = {}, acc1 = {}, acc2 = {}, acc3 = {};

  if (wave == 0) {                       // prologue: DMA chunk 0 into buffer 0
    tdm_load_2d(as_base, A + (size_t)m0 * lda, KSTEP, MT, (uint64_t)lda, A_PAD_BITS);
    tdm_load_2d(bs_base, B, NF, KSTEP, NF, 0u);
  }

  for (int i = 0; i < nchunks; ++i) {
    const int buf = i & 1;
    if (wave == 0) {
      if (i + 1 < nchunks) {             // DMA next chunk into the other buffer
        const int k1 = (i + 1) * KSTEP;
        tdm_load_2d(as_base + (buf ^ 1) * (MT * AP * 4),
                    A + (size_t)m0 * lda + k1, KSTEP, MT, (uint64_t)lda, A_PAD_BITS);
        tdm_load_2d(bs_base + (buf ^ 1) * (KSTEP * NF * 4),
                    B + (size_t)k1 * NF, NF, KSTEP, NF, 0u);
        __builtin_amdgcn_s_wait_tensorcnt(1);   // A+B of chunk i complete (in-order)
      } else {
        __builtin_amdgcn_s_wait_tensorcnt(0);
      }
    }
    __syncthreads();

    const float* Ab = &As[buf * (MT * AP)];
    const float* Bb = &Bs[buf * (KSTEP * NF)];
#pragma unroll
    for (int s = 0; s < KSTEP / 4; ++s) {
      // A fragment 16x4: lanes 0-15 -> K{0,1}, lanes 16-31 -> K{2,3}
      v2f af = *(const v2f*)&Ab[(mt * 16 + l15) * AP + s * 4 + 2 * khalf];
      const int kk = s * 4 + 2 * khalf;
      const float* brow = &Bb[kk * NF + l15];
      {
        v2f bf; bf.x = brow[(ntb + 0) * 16]; bf.y = brow[NF + (ntb + 0) * 16];
        acc0 = __builtin_amdgcn_wmma_f32_16x16x4_f32(false, af, false, bf,
                                                     (short)0, acc0, false, false);
      }
      {
        v2f bf; bf.x = brow[(ntb + 1) * 16]; bf.y = brow[NF + (ntb + 1) * 16];
        acc1 = __builtin_amdgcn_wmma_f32_16x16x4_f32(false, af, false, bf,
                                                     (short)0, acc1, false, false);
      }
      {
        v2f bf; bf.x = brow[(ntb + 2) * 16]; bf.y = brow[NF + (ntb + 2) * 16];
        acc2 = __builtin_amdgcn_wmma_f32_16x16x4_f32(false, af, false, bf,
                                                     (short)0, acc2, false, false);
      }
      {
        v2f bf; bf.x = brow[(ntb + 3) * 16]; bf.y = brow[NF + (ntb + 3) * 16];
        acc3 = __builtin_amdgcn_wmma_f32_16x16x4_f32(false, af, false, bf,
                                                     (short)0, acc3, false, false);
      }
    }
    __syncthreads();   // done reading buf before TDM overwrites it (chunk i+2)
  }

  // Epilogue. C/D layout: VGPR v, lanes 0-15 -> M=v, lanes 16-31 -> M=v+8.
#pragma unroll
  for (int v = 0; v < 8; ++v) {
    const int row = m0 + mt * 16 + v + 8 * khalf;
    float scale = 1.0f;
    if (mode == 1) scale = -dinv[row];
    else if (mode == 2) scale = -2.0f * dinv[row];
    float vals[4] = {acc0[v], acc1[v], acc2[v], acc3[v]};
#pragma unroll
    for (int ti = 0; ti < 4; ++ti) {
      const int col = (ntb + ti) * 16 + l15;
      const size_t idx = (size_t)row * NF + col;
      float val = vals[ti];
      if (mode == 0) {
        out[idx] = val;
      } else if (mode == 1) {
        out[idx] = scale * val;
      } else if (mode == 2) {
        out[idx] = scale * val - xorig[idx];
      } else { // mode 3: accumulate
        out[idx] += val;
      }
    }
  }
}

// ---------------------------------------------------------------------------
// Kernel 5: out = a*out + (1-a)*0.001*cheb + bias   (out already holds gcn)
// ---------------------------------------------------------------------------
__global__ void blend_kernel(float* __restrict__ out,
                             const float* __restrict__ cheb,
                             const float* __restrict__ alpha,
                             const float* __restrict__ bias) {
  const size_t i4 = (size_t)blockIdx.x * blockDim.x + threadIdx.x;
  const float a = 1.0f / (1.0f + expf(-alpha[0]));
  const float b1 = (1.0f - a) * 0.001f;
  float4 g = ((const float4*)out)[i4];
  float4 c = ((const float4*)cheb)[i4];
  const int col = (int)((i4 * 4) & (NF - 1));
  const float4 bs = *(const float4*)(bias + col);
  g.x = a * g.x + b1 * c.x + bs.x;
  g.y = a * g.y + b1 * c.y + bs.y;
  g.z = a * g.z + b1 * c.z + bs.z;
  g.w = a * g.w + b1 * c.w + bs.w;
  ((float4*)out)[i4] = g;
}

// ---------------------------------------------------------------------------
extern "C" void kernel_launch(void* const* d_in, const int* in_sizes, int n_in,
                              void* d_out, int out_size, void* d_ws, size_t ws_size,
                              hipStream_t stream) {
  (void)in_sizes; (void)n_in; (void)out_size; (void)ws_size;
  const float* x     = (const float*)d_in[0];   // [8192,256]
  const float* adj   = (const float*)d_in[1];   // [8192,8192]
  const float* gw    = (const float*)d_in[2];   // [256,256]
  const float* cw    = (const float*)d_in[3];   // [3,256,256]
  const float* alpha = (const float*)d_in[4];   // scalar
  const float* bias  = (const float*)d_in[5];   // [256]
  float* out = (float*)d_out;                   // [8192,256]
  float* ws  = (float*)d_ws;

  // workspace layout (floats); support/cheb aliased, xs reused for t1s
  float* dinv    = ws;                  // 8192
  float* Wn      = ws + 8192;           // 3*256*256 = 196608
  float* support = ws + 204800;         // 2097152  (later reused as cheb acc)
  float* xs      = ws + 2301952;        // 2097152  (xs, then t1s)
  float* T1      = ws + 4399104;        // 2097152
  float* T2      = ws + 6496256;        // 2097152  -> total 8593408 floats (~34.4 MB)
  float* cheb    = support;

  const int gemm_blocks = NROWS / MT;             // 256
  const int ew_blocks   = (NROWS * NF / 4) / 256; // 2048

  // weight normalization + degrees
  prep_weights_kernel<<<3, 256, 0, stream>>>(cw, Wn);
  degree_kernel<<<NROWS, 256, 0, stream>>>(adj, dinv);

  // GCN branch: support = x @ gw ; out = adj @ support
  gemm_wmma_kernel<<<gemm_blocks, 256, 0, stream>>>(x, NF, NF, gw, support,
                                                    nullptr, nullptr, 0);
  gemm_wmma_kernel<<<gemm_blocks, 256, 0, stream>>>(adj, NROWS, NROWS, support, out,
                                                    nullptr, nullptr, 0);

  // Chebyshev basis: T1 = -D^{-1/2} (adj @ (D^{-1/2} x)) ; T2 = 2 L_hat T1 - x
  scale_rows_kernel<<<ew_blocks, 256, 0, stream>>>(x, dinv, xs);
  gemm_wmma_kernel<<<gemm_blocks, 256, 0, stream>>>(adj, NROWS, NROWS, xs, T1,
                                                    dinv, nullptr, 1);
  scale_rows_kernel<<<ew_blocks, 256, 0, stream>>>(T1, dinv, xs);   // t1s
  gemm_wmma_kernel<<<gemm_blocks, 256, 0, stream>>>(adj, NROWS, NROWS, xs, T2,
                                                    dinv, x, 2);

  // cheb = x@Wn0 + T1@Wn1 + T2@Wn2   (support buffer is free now)
  gemm_wmma_kernel<<<gemm_blocks, 256, 0, stream>>>(x, NF, NF, Wn, cheb,
                                                    nullptr, nullptr, 0);
  gemm_wmma_kernel<<<gemm_blocks, 256, 0, stream>>>(T1, NF, NF, Wn + NF * NF, cheb,
                                                    nullptr, nullptr, 3);
  gemm_wmma_kernel<<<gemm_blocks, 256, 0, stream>>>(T2, NF, NF, Wn + 2 * NF * NF, cheb,
                                                    nullptr, nullptr, 3);

  // final blend into d_out
  blend_kernel<<<ew_blocks, 256, 0, stream>>>(out, cheb, alpha, bias);
}